// Pin2PinAttraction_85117661872810
// MI455X (gfx1250) — compile-verified
//
#include <hip/hip_runtime.h>

// Native clang ext-vector types (required by __builtin_nontemporal_load and WMMA)
typedef __attribute__((ext_vector_type(2))) float v2f;
typedef __attribute__((ext_vector_type(8))) float v8f;
typedef __attribute__((ext_vector_type(2))) unsigned long long v2u64;

#define NBLOCKS 2048
#define TPB 256

// Pass 1: per-pair weighted squared distance, block-level partial sums.
// Streams pairs/weights with non-temporal hints (keep 16MB pin_pos resident in
// the 192MB L2); gathers use default RT caching.
__global__ void pair_partial_kernel(const float* __restrict__ x,
                                    const float* __restrict__ y,
                                    const float* __restrict__ w,
                                    const v2u64* __restrict__ pairs,
                                    float* __restrict__ partials,
                                    int num_pairs) {
    const int tid    = blockIdx.x * blockDim.x + threadIdx.x;
    const int stride = gridDim.x * blockDim.x;

    float acc = 0.0f;
    for (int i = tid; i < num_pairs; i += stride) {
        // prefetch the next stream chunk (global_prefetch_b8, speculative)
        __builtin_prefetch(&pairs[i + stride], 0, 0);
        __builtin_prefetch(&w[i + stride], 0, 0);

        v2u64 ab = __builtin_nontemporal_load(&pairs[i]);   // b128, NT
        float wt = __builtin_nontemporal_load(&w[i]);       // b32,  NT
        int a = (int)ab.x;
        int b = (int)ab.y;
        float dx = x[a] - x[b];
        float dy = y[a] - y[b];
        acc = fmaf(wt, fmaf(dx, dx, dy * dy), acc);
    }

    // wave32 reduction
    #pragma unroll
    for (int off = 16; off > 0; off >>= 1)
        acc += __shfl_down(acc, off, 32);

    __shared__ float smem[TPB / 32];
    const int lane = threadIdx.x & 31;
    const int wid  = threadIdx.x >> 5;
    if (lane == 0) smem[wid] = acc;
    __syncthreads();

    if (threadIdx.x == 0) {
        float s = 0.0f;
        #pragma unroll
        for (int i = 0; i < TPB / 32; ++i) s += smem[i];
        partials[blockIdx.x] = s;
    }
}

// Pass 2: single wave (32 lanes, EXEC all ones). Each lane strided-sums the
// block partials, then a V_WMMA_F32_16X16X4_F32 with B=ones folds the 32 lane
// values: A[m][0]=lane m, A[m][2]=lane m+16 (K=1,3 zero), so
// D[m][n] = lane_m + lane_{m+16}. Row-sums live in c[0..7] of lane 0 (M=0..7)
// and lane 16 (M=8..15); one shuffle finishes the scalar. Deterministic.
__global__ void final_reduce_wmma_kernel(const float* __restrict__ partials,
                                         float* __restrict__ out,
                                         int n) {
    const int lane = threadIdx.x;  // blockDim.x == 32, one full wave
    float acc = 0.0f;
    for (int i = lane; i < n; i += 32) acc += partials[i];

    v2f a;
    a.x = acc;   // K=0 (lanes 0-15) / K=2 (lanes 16-31)
    a.y = 0.0f;  // K=1 / K=3
    v2f b;
    b.x = 1.0f;  // B rows K=0 / K=2, all columns
    b.y = 1.0f;  // B rows K=1 / K=3
    v8f c = {};
    c = __builtin_amdgcn_wmma_f32_16x16x4_f32(
        /*neg_a=*/false, a, /*neg_b=*/false, b,
        /*c_mod=*/(short)0, c, /*reuse_a=*/false, /*reuse_b=*/false);

    float s = c[0] + c[1] + c[2] + c[3] + c[4] + c[5] + c[6] + c[7];
    s += __shfl_down(s, 16, 32);  // lane0: rows 0-7 + rows 8-15
    if (lane == 0) out[0] = s;
}

extern "C" void kernel_launch(void* const* d_in, const int* in_sizes, int n_in,
                              void* d_out, int out_size, void* d_ws, size_t ws_size,
                              hipStream_t stream) {
    // setup_inputs order: pin_pos(f32, 2*NUM_PINS), weights(f32, NUM_PAIRS),
    //                     pairs(int64, 2*NUM_PAIRS), pin_mask(bool, unused)
    const float* pin_pos = (const float*)d_in[0];
    const float* weights = (const float*)d_in[1];
    const v2u64* pairs   = (const v2u64*)d_in[2];

    const int num_pins  = in_sizes[0] / 2;
    const int num_pairs = in_sizes[1];

    const float* x = pin_pos;
    const float* y = pin_pos + num_pins;
    float* partials = (float*)d_ws;  // NBLOCKS floats (8 KB)

    pair_partial_kernel<<<NBLOCKS, TPB, 0, stream>>>(
        x, y, weights, pairs, partials, num_pairs);
    final_reduce_wmma_kernel<<<1, 32, 0, stream>>>(
        partials, (float*)d_out, NBLOCKS);
}